// PropagateLayer_17892833755266
// MI455X (gfx1250) — compile-verified
//
#include <hip/hip_runtime.h>
#include <math.h>

// Problem constants (match reference)
#define NN 50000
#define IND 128      // in_dim = H*F
#define NH 4         // heads
#define NF 32        // feats per head
#define NE 500000    // edges per type
#define NL 512       // label dim
#define SLOPE 0.2f

typedef __attribute__((ext_vector_type(2))) float v2f;
typedef __attribute__((ext_vector_type(8))) float v8f;

// ---- float <-> monotone-uint mapping for atomic segment-max on f32 ----
__device__ __forceinline__ unsigned int f2key(float x) {
  unsigned int b = __float_as_uint(x);
  return b ^ ((unsigned int)((int)b >> 31) | 0x80000000u);
}
__device__ __forceinline__ float key2f(unsigned int k) {
  unsigned int b = (k & 0x80000000u) ? (k ^ 0x80000000u) : ~k;
  return __uint_as_float(b);
}

// ============================================================================
// GEMM: fs = x @ Wsrc^T + bsrc ; fd = x @ Wdst^T + bdst  via V_WMMA_F32_16X16X4_F32
// One wave = one 16x16 output tile of one of the two outputs.
// A (16x4 f32): lane<16 -> row=lane, {K0,K1}; lane>=16 -> row=lane-16, {K2,K3}
// B (4x16 f32): VGPR0 holds K = 0/2 (lo/hi half-wave), VGPR1 holds K = 1/3; col = lane%16
// C/D (16x16 f32): VGPR r -> row r (+8 for hi half), col = lane%16
// ============================================================================
__global__ void gemm_wmma_kernel(const float* __restrict__ x,
                                 const float* __restrict__ Wsrc, const float* __restrict__ bsrc,
                                 const float* __restrict__ Wdst, const float* __restrict__ bdst,
                                 float* __restrict__ fs, float* __restrict__ fd) {
  const int wave  = (blockIdx.x * blockDim.x + threadIdx.x) >> 5;
  const int lane  = threadIdx.x & 31;
  const int which = wave & 1;               // 0 -> src weights, 1 -> dst weights
  const int colT  = (wave >> 1) & 7;        // 8 column tiles of 16
  const int rowT  = wave >> 4;              // 3125 row tiles of 16
  if (rowT >= NN / 16) return;

  const float* __restrict__ W    = which ? Wdst : Wsrc;
  const float* __restrict__ bias = which ? bdst : bsrc;
  float* __restrict__ out        = which ? fd : fs;

  const int hi   = lane >> 4;               // half-wave select
  const int lm   = lane & 15;
  const int arow = rowT * 16 + lm;          // A-matrix row handled by this lane
  const int ocol = colT * 16 + lm;          // output / B column handled by this lane

  v8f c = {};
  #pragma unroll 4
  for (int k = 0; k < IND / 4; ++k) {
    const int kk = k * 4 + hi * 2;          // identical K slot layout for A and B halves
    v2f a = *(const v2f*)(x + arow * IND + kk);          // A[arow][kk], A[arow][kk+1]
    v2f b = *(const v2f*)(W + ocol * IND + kk);          // B[kk][ocol] = W[ocol][kk]
    c = __builtin_amdgcn_wmma_f32_16x16x4_f32(false, a, false, b,
                                              (short)0, c, false, false);
  }
  const float bb = bias[ocol];
  #pragma unroll
  for (int r = 0; r < 8; ++r) {
    const int row = rowT * 16 + r + hi * 8;
    out[row * IND + ocol] = c[r] + bb;
  }
}

// ============================================================================
// Per-edge logits: logit[e,h] = sum_f leaky(fs[src]+fd[dst]) * attn[h,f]
// One wave per edge; lane covers 4 contiguous feats (head = lane/8); also
// atomicMax segment-max per (dst, head) in monotone-uint space.
// ============================================================================
__global__ void logits_kernel(const float* __restrict__ fs, const float* __restrict__ fd,
                              const float* __restrict__ attn,
                              const int* __restrict__ src, const int* __restrict__ dst,
                              float* __restrict__ logits, unsigned int* __restrict__ mkey) {
  const int e    = (blockIdx.x * blockDim.x + threadIdx.x) >> 5;
  const int lane = threadIdx.x & 31;
  if (e >= NE) return;
  const int s = src[e], d = dst[e];
  const float4 fsv = *(const float4*)(fs + (size_t)s * IND + lane * 4);
  const float4 fdv = *(const float4*)(fd + (size_t)d * IND + lane * 4);
  const float4 av  = *(const float4*)(attn + lane * 4);
  float t, sum = 0.f;
  t = fsv.x + fdv.x; sum += (t > 0.f ? t : SLOPE * t) * av.x;
  t = fsv.y + fdv.y; sum += (t > 0.f ? t : SLOPE * t) * av.y;
  t = fsv.z + fdv.z; sum += (t > 0.f ? t : SLOPE * t) * av.z;
  t = fsv.w + fdv.w; sum += (t > 0.f ? t : SLOPE * t) * av.w;
  // reduce within 8-lane groups (one head per group)
  sum += __shfl_xor(sum, 1, 32);
  sum += __shfl_xor(sum, 2, 32);
  sum += __shfl_xor(sum, 4, 32);
  if ((lane & 7) == 0) {
    const int h = lane >> 3;
    logits[e * NH + h] = sum;
    atomicMax(&mkey[d * NH + h], f2key(sum));
  }
}

// ============================================================================
// ex = exp(logit - max[dst]); segment-sum per (dst,head). One thread per (e,h).
// ============================================================================
__global__ void expsum_kernel(const int* __restrict__ dst,
                              const unsigned int* __restrict__ mkey,
                              float* __restrict__ logits_ex,
                              float* __restrict__ ssum) {
  const int i = blockIdx.x * blockDim.x + threadIdx.x;
  if (i >= NE * NH) return;
  const int e = i >> 2, h = i & 3;
  const int d = dst[e];
  const float m = key2f(mkey[d * NH + h]);
  const float ex = __expf(logits_ex[i] - m);
  logits_ex[i] = ex;
  atomicAdd(&ssum[d * NH + h], ex);
}

// ============================================================================
// a = ex / s[dst]; a_mean = mean_h(a); ft[dst] += fs[src] * a   (wave per edge)
// ============================================================================
__global__ void scatter_kernel(const float* __restrict__ fs,
                               const int* __restrict__ src, const int* __restrict__ dst,
                               const float* __restrict__ ex, const float* __restrict__ ssum,
                               float* __restrict__ ft, float* __restrict__ amean) {
  const int e    = (blockIdx.x * blockDim.x + threadIdx.x) >> 5;
  const int lane = threadIdx.x & 31;
  if (e >= NE) return;
  const int s = src[e], d = dst[e];
  const float4 exv = *(const float4*)(ex + (size_t)e * NH);
  const float4 sv  = *(const float4*)(ssum + (size_t)d * NH);
  float4 a;
  a.x = exv.x / sv.x; a.y = exv.y / sv.y; a.z = exv.z / sv.z; a.w = exv.w / sv.w;
  if (lane == 0) amean[e] = 0.25f * (a.x + a.y + a.z + a.w);
  const float ah = (lane < 8) ? a.x : (lane < 16) ? a.y : (lane < 24) ? a.z : a.w;
  const float4 fsv = *(const float4*)(fs + (size_t)s * IND + lane * 4);
  float* base = ft + (size_t)d * IND + lane * 4;
  atomicAdd(base + 0, fsv.x * ah);
  atomicAdd(base + 1, fsv.y * ah);
  atomicAdd(base + 2, fsv.z * ah);
  atomicAdd(base + 3, fsv.w * ah);
}

// ============================================================================
// Label propagation: yh[dst] += y[src] * a_mean[e]   (128 threads x float4 per edge)
// ============================================================================
__global__ void label_kernel(const float* __restrict__ y,
                             const int* __restrict__ src, const int* __restrict__ dst,
                             const float* __restrict__ amean, float* __restrict__ yh) {
  const int i = blockIdx.x * blockDim.x + threadIdx.x;   // < NE * 128
  const int e = i >> 7, t = i & 127;
  if (e >= NE) return;
  const int s = src[e], d = dst[e];
  const float am = amean[e];
  const float4 yv = *(const float4*)(y + (size_t)s * NL + t * 4);
  float* base = yh + (size_t)d * NL + t * 4;
  atomicAdd(base + 0, yv.x * am);
  atomicAdd(base + 1, yv.y * am);
  atomicAdd(base + 2, yv.z * am);
  atomicAdd(base + 3, yv.w * am);
}

// ============================================================================
// h = elu( mean_h( ft_p+ft_s + 2x ) )   (ft holds both types accumulated)
// ============================================================================
__global__ void hout_kernel(const float* __restrict__ ft, const float* __restrict__ x,
                            float* __restrict__ hout) {
  const int i = blockIdx.x * blockDim.x + threadIdx.x;   // n*NF + f
  if (i >= NN * NF) return;
  const int n = i >> 5, f = i & 31;
  float acc = 0.f;
  #pragma unroll
  for (int hh = 0; hh < NH; ++hh) {
    const int idx = n * IND + hh * NF + f;
    acc += ft[idx] + 2.f * x[idx];
  }
  acc *= 0.25f;
  hout[i] = acc > 0.f ? acc : (__expf(acc) - 1.f);
}

// ============================================================================
// Row-wise L2 normalize yh (in place in d_out), override with y where dst_flag.
// One wave per node; 16 elements per lane.
// ============================================================================
__global__ void norm_kernel(const float* __restrict__ y,
                            const unsigned char* __restrict__ flag,
                            float* __restrict__ yhat) {
  const int n    = (blockIdx.x * blockDim.x + threadIdx.x) >> 5;
  const int lane = threadIdx.x & 31;
  if (n >= NN) return;
  float* row = yhat + (size_t)n * NL;
  float4 v[4];
  float ss = 0.f;
  #pragma unroll
  for (int j = 0; j < 4; ++j) {
    v[j] = *(const float4*)(row + lane * 4 + j * 128);
    ss += v[j].x * v[j].x + v[j].y * v[j].y + v[j].z * v[j].z + v[j].w * v[j].w;
  }
  #pragma unroll
  for (int m = 16; m >= 1; m >>= 1) ss += __shfl_xor(ss, m, 32);
  const float inv = 1.f / fmaxf(sqrtf(ss), 1e-12f);
  const bool keep = flag[n] != 0;
  const float* yrow = y + (size_t)n * NL;
  #pragma unroll
  for (int j = 0; j < 4; ++j) {
    float4 o;
    if (keep) {
      o = *(const float4*)(yrow + lane * 4 + j * 128);
    } else {
      o.x = v[j].x * inv; o.y = v[j].y * inv; o.z = v[j].z * inv; o.w = v[j].w * inv;
    }
    *(float4*)(row + lane * 4 + j * 128) = o;
  }
}

// ============================================================================
extern "C" void kernel_launch(void* const* d_in, const int* in_sizes, int n_in,
                              void* d_out, int out_size, void* d_ws, size_t ws_size,
                              hipStream_t stream) {
  const float* x    = (const float*)d_in[0];
  const float* y    = (const float*)d_in[1];
  const float* Wsrc = (const float*)d_in[2];
  const float* bsrc = (const float*)d_in[3];
  const float* Wdst = (const float*)d_in[4];
  const float* bdst = (const float*)d_in[5];
  const float* attn = (const float*)d_in[6];
  const int* src_p  = (const int*)d_in[7];
  const int* dst_p  = (const int*)d_in[8];
  const int* src_s  = (const int*)d_in[9];
  const int* dst_s  = (const int*)d_in[10];
  const unsigned char* dflag = (const unsigned char*)d_in[11];  // jax bool = 1 byte

  // workspace layout (~91 MB)
  float* fs            = (float*)d_ws;                    // N*128
  float* fd            = fs + (size_t)NN * IND;           // N*128
  float* ft            = fd + (size_t)NN * IND;           // N*128 (both types accumulate)
  float* ex            = ft + (size_t)NN * IND;           // E*4  (logits -> ex, reused per type)
  unsigned int* mkey   = (unsigned int*)(ex + (size_t)NE * NH);  // N*4
  float* ssum          = (float*)(mkey + (size_t)NN * NH);       // N*4
  float* amean_p       = ssum + (size_t)NN * NH;          // E
  float* amean_s       = amean_p + NE;                    // E

  float* hout = (float*)d_out;                 // N*32
  float* yhat = hout + (size_t)NN * NF;        // N*512 (yh accumulated in place)

  // zero-init accumulators (hipMemsetAsync is graph-capture safe)
  hipMemsetAsync(ft,   0, sizeof(float) * (size_t)NN * IND, stream);
  hipMemsetAsync(yhat, 0, sizeof(float) * (size_t)NN * NL, stream);

  // shared GEMMs (fs, fd used by both edge types)
  gemm_wmma_kernel<<<(NN / 16) * 8 * 2 / 8, 256, 0, stream>>>(x, Wsrc, bsrc, Wdst, bdst, fs, fd);

  // ---- edge type p ----
  hipMemsetAsync(mkey, 0, sizeof(unsigned int) * (size_t)NN * NH, stream);
  hipMemsetAsync(ssum, 0, sizeof(float) * (size_t)NN * NH, stream);
  logits_kernel<<<NE / 8, 256, 0, stream>>>(fs, fd, attn, src_p, dst_p, ex, mkey);
  expsum_kernel<<<(NE * NH + 255) / 256, 256, 0, stream>>>(dst_p, mkey, ex, ssum);
  scatter_kernel<<<NE / 8, 256, 0, stream>>>(fs, src_p, dst_p, ex, ssum, ft, amean_p);

  // ---- edge type s ----
  hipMemsetAsync(mkey, 0, sizeof(unsigned int) * (size_t)NN * NH, stream);
  hipMemsetAsync(ssum, 0, sizeof(float) * (size_t)NN * NH, stream);
  logits_kernel<<<NE / 8, 256, 0, stream>>>(fs, fd, attn, src_s, dst_s, ex, mkey);
  expsum_kernel<<<(NE * NH + 255) / 256, 256, 0, stream>>>(dst_s, mkey, ex, ssum);
  scatter_kernel<<<NE / 8, 256, 0, stream>>>(fs, src_s, dst_s, ex, ssum, ft, amean_s);

  // ---- label propagation (both types into yhat) ----
  label_kernel<<<NE * (NL / 4) / 256, 256, 0, stream>>>(y, src_p, dst_p, amean_p, yhat);
  label_kernel<<<NE * (NL / 4) / 256, 256, 0, stream>>>(y, src_s, dst_s, amean_s, yhat);

  // ---- outputs ----
  hout_kernel<<<(NN * NF + 255) / 256, 256, 0, stream>>>(ft, x, hout);
  norm_kernel<<<NN / 8, 256, 0, stream>>>(y, dflag, yhat);
}